// HashZchWriteSparseArch_17282948399338
// MI455X (gfx1250) — compile-verified
//
#include <hip/hip_runtime.h>
#include <cstdint>

// HashZch remap + embedding gather for MI455X (gfx1250, wave32).
// Pure data movement: ~1.05 GB total traffic -> ~45us floor @ 23.3 TB/s.
// No FLOPs => no WMMA; optimize the vmem path: b128 coalesced gather with
// scalar (SADDR) row bases via v_readlane, NT streamed stores, prefetch,
// L2-resident probe table.

typedef float v4f __attribute__((ext_vector_type(4)));

constexpr int kZch        = 2000000;          // table.num_embeddings
constexpr int kBuckets    = 4;                // total_num_buckets (pow2)
constexpr int kBucketSize = kZch / kBuckets;  // 500000
constexpr int kProbe      = 5;                // max_probe
constexpr int kDim        = 128;              // embedding_dim

__device__ __forceinline__ uint32_t mix32(uint32_t x) {
  x ^= x >> 16;
  x *= 0x7FEB352Du;
  x ^= x >> 15;
  x *= 0x846CA68Bu;
  x ^= x >> 16;
  return x;
}

__global__ __launch_bounds__(256) void hashzch_remap_gather(
    const int* __restrict__ ids, const int* __restrict__ identity,
    const float* __restrict__ emb, float* __restrict__ out_val,
    float* __restrict__ out_remap, int n) {
  const int lane   = (int)(threadIdx.x & 31u);
  const int gtid   = (int)(blockIdx.x * blockDim.x + threadIdx.x);
  const int idBase = gtid & ~31;            // wave-uniform base id
  if (idBase >= n) return;
  const int  idx   = idBase + lane;
  const bool valid = idx < n;

  // ---------- phase 1: hash + linear probe, one id per lane ----------
  const int      my_id  = valid ? ids[idx] : 0;
  const uint32_t h      = mix32((uint32_t)my_id);
  const uint32_t bucket = h & (uint32_t)(kBuckets - 1);
  const uint32_t slot   = (h >> 2) % (uint32_t)kBucketSize;
  const uint32_t base   = bucket * (uint32_t)kBucketSize;

  int cand[kProbe];
  int probe[kProbe];
#pragma unroll
  for (int p = 0; p < kProbe; ++p) {
    uint32_t s = slot + (uint32_t)p;          // slot+p < bucketSize+5: cheap mod
    if (s >= (uint32_t)kBucketSize) s -= (uint32_t)kBucketSize;
    cand[p]  = (int)(base + s);
    probe[p] = identity[cand[p]];             // 8MB table: L2-resident, default RT
  }
  int remap = cand[0];                        // miss -> base slot
#pragma unroll
  for (int p = kProbe - 1; p >= 0; --p)       // reverse scan => first hit wins
    if (probe[p] == my_id) remap = cand[p];

  if (valid) {
    // remapped slot ids < 2e6 < 2^24: exact in fp32 (d_out is a float buffer)
    __builtin_nontemporal_store((float)remap, out_remap + idx);
    // Warm this lane's embedding row before the wave-serial copy loop
    // consumes it (lowers to global_prefetch_b8; row = 512B).
    const float* row = emb + (size_t)remap * kDim;
    __builtin_prefetch(row, 0, 1);
    __builtin_prefetch(row + 64, 0, 1);
  }

  // ---------- phase 2: gather rows, one row per readlane-broadcast slot ----
  // v_readlane puts the slot in an SGPR -> row base is a scalar address and
  // each row is one SADDR-form global_load_b128 per lane (128 floats =
  // 32 lanes x float4 = 512B, fully coalesced). No LDS traffic at all.
  const int col    = lane << 2;
  float*    outRow = out_val + (size_t)idBase * kDim + col;
  const int cnt    = (n - idBase >= 32) ? 32 : (n - idBase);

  if (cnt == 32) {
#pragma unroll
    for (int j = 0; j < 32; j += 4) {
      // 4 independent rows in flight: loads cluster before the first wait.
      const int r0 = __builtin_amdgcn_readlane(remap, j + 0);
      const int r1 = __builtin_amdgcn_readlane(remap, j + 1);
      const int r2 = __builtin_amdgcn_readlane(remap, j + 2);
      const int r3 = __builtin_amdgcn_readlane(remap, j + 3);
      const v4f v0 = *(const v4f*)(emb + (size_t)r0 * kDim + col);
      const v4f v1 = *(const v4f*)(emb + (size_t)r1 * kDim + col);
      const v4f v2 = *(const v4f*)(emb + (size_t)r2 * kDim + col);
      const v4f v3 = *(const v4f*)(emb + (size_t)r3 * kDim + col);
      // Streamed 516MB output: non-temporal so it never evicts the
      // L2-resident identity table / reusable emb lines.
      __builtin_nontemporal_store(v0, (v4f*)(outRow + (size_t)(j + 0) * kDim));
      __builtin_nontemporal_store(v1, (v4f*)(outRow + (size_t)(j + 1) * kDim));
      __builtin_nontemporal_store(v2, (v4f*)(outRow + (size_t)(j + 2) * kDim));
      __builtin_nontemporal_store(v3, (v4f*)(outRow + (size_t)(j + 3) * kDim));
    }
  } else {
    for (int j = 0; j < cnt; ++j) {          // j is wave-uniform
      const int r = __builtin_amdgcn_readlane(remap, j);
      const v4f v = *(const v4f*)(emb + (size_t)r * kDim + col);
      __builtin_nontemporal_store(v, (v4f*)(outRow + (size_t)j * kDim));
    }
  }
}

extern "C" void kernel_launch(void* const* d_in, const int* in_sizes, int n_in,
                              void* d_out, int out_size, void* d_ws, size_t ws_size,
                              hipStream_t stream) {
  const int*   ids      = (const int*)d_in[0];
  const int*   identity = (const int*)d_in[1];
  const float* emb      = (const float*)d_in[2];
  const int    n        = in_sizes[0];

  float* out_val   = (float*)d_out;                  // [n, 128] gathered rows
  float* out_remap = out_val + (size_t)n * kDim;     // [n] remapped slots (fp32-exact)

  const int threads = 256;                            // 8 wave32 per block
  const int blocks  = (n + threads - 1) / threads;    // 1 id per thread
  hipLaunchKernelGGL(hashzch_remap_gather, dim3(blocks), dim3(threads), 0, stream,
                     ids, identity, emb, out_val, out_remap, n);
}